// RandomWindowWarping_69217692942481
// MI455X (gfx1250) — compile-verified
//
#include <hip/hip_runtime.h>
#include <cstdint>

// Window-warp resampling, fused single pass.
// Memory-bound: >=256MB traffic -> ~11us floor @ 23.3TB/s. No matmul
// structure, so the CDNA5 feature used is the async global->LDS copy engine
// (ASYNCcnt): each input byte is read from HBM exactly once via coalesced
// 128-bit global_load_async_to_lds_b128, and the irregular interpolation
// gathers are served from LDS instead of VMEM.

namespace {
constexpr int kB = 128;
constexpr int kC = 16;
constexpr int kT = 16384;
constexpr int kW = 4096;          // warp window = 0.25 * T
constexpr int kTile = 256;        // outputs (t values) per block
constexpr int kTilesPerB = kT / kTile;  // 64
// Max source span per tile: shrink case slope = (4095/2047)*(L-1)/(T-1)
//   -> span <= 2.0005 * 225 + 2 ~= 452; +3 align-down, +3 round-up => <= 460.
constexpr int kSrcCap = 480;      // multiple of 4
constexpr int kSrcStride = 488;   // per-channel LDS row stride (floats),
                                  // multiple of 4 -> 16B-aligned rows
}  // namespace

// Stage-1 warp map: warped index j -> fractional source position in x.
// Matches reference fp32 op order.
__device__ __forceinline__ float warp_pos(float jf, float sf, float nlf) {
  const float send = sf + nlf;
  float pos;
  if (jf >= sf && jf < send) {
    const float denom = fmaxf(nlf - 1.0f, 1.0f);
    pos = sf + ((jf - sf) * (float)(kW - 1)) / denom;
  } else if (jf >= send) {
    pos = (jf - nlf) + (float)kW;
  } else {
    pos = jf;
  }
  return fminf(fmaxf(pos, 0.0f), (float)(kT - 1));
}

__global__ __launch_bounds__(kTile) void window_warp_kernel(
    const float* __restrict__ x, const int* __restrict__ win_start,
    const int* __restrict__ new_len, float* __restrict__ out) {
  __shared__ __align__(16) float sbuf[kC * kSrcStride];

  const int blk = blockIdx.x;
  const int b = blk / kTilesPerB;
  const int tile = blk % kTilesPerB;
  const int tid = threadIdx.x;

  const int s = win_start[b];    // scalar (uniform) loads
  const int nl = new_len[b];
  const int L = kT + nl - kW;    // true warped length
  const float sf = (float)s;
  const float nlf = (float)nl;
  const float Lm1 = (float)(L - 1);

  const int t0 = tile * kTile;

  // ---- Block-uniform conservative source range in x-time for this tile ----
  // Both maps are monotone non-decreasing (also under fp eval: mul/div by a
  // positive constant is weakly monotone), so endpoint evaluation bounds all
  // threads' gather indices.
  float q0 = ((float)t0 * Lm1) / (float)(kT - 1);
  int j0 = (int)floorf(q0);
  j0 = j0 < 0 ? 0 : j0;
  int lo_raw = (int)floorf(warp_pos((float)j0, sf, nlf));
  float q1 = ((float)(t0 + kTile - 1) * Lm1) / (float)(kT - 1);
  int j1 = (int)floorf(q1) + 1;
  j1 = j1 > (L - 1) ? (L - 1) : j1;
  int hi_max = (int)floorf(warp_pos((float)j1, sf, nlf)) + 1;
  hi_max = hi_max > (kT - 1) ? (kT - 1) : hi_max;

  // Align the staged window to 16B on both ends for b128 async copies.
  const int lo_min = lo_raw & ~3;
  int count = hi_max - lo_min + 1;
  count = (count + 3) & ~3;
  count = count < 4 ? 4 : (count > kSrcCap ? kSrcCap : count);

  // ---- Async-copy x[b, :, lo_min : lo_min+count] into LDS (b128/lane) ----
  // Per-lane LDS byte address = low 32 bits of the generic shared pointer
  // (LDS aperture keeps the offset in addr[31:0]).
  const int n4 = count >> 2;  // 16B packets per channel row (<=120)
  const float* xb = x + (size_t)b * kC * kT + lo_min;
#pragma unroll
  for (int c = 0; c < kC; ++c) {
    const float* src = xb + (size_t)c * kT;
    float* dst = &sbuf[c * kSrcStride];
    for (int i4 = tid; i4 < n4; i4 += kTile) {  // <=1 masked iteration
      unsigned lds_addr = (unsigned)(uintptr_t)(dst + 4 * i4);
      unsigned long long gaddr = (unsigned long long)(uintptr_t)(src + 4 * i4);
      asm volatile("global_load_async_to_lds_b128 %0, %1, off" ::"v"(lds_addr),
                   "v"(gaddr)
                   : "memory");
    }
  }
  asm volatile("s_wait_asynccnt 0x0" ::: "memory");
  __syncthreads();

  // ---- Compute one output t per thread; positions shared by all channels ----
  const int t = t0 + tid;
  const float tf = (float)t;
  const float q = (tf * Lm1) / (float)(kT - 1);
  const float qfl = floorf(q);
  const float qf = q - qfl;  // reference uses unclamped floor for the fraction
  int qlo = (int)qfl;
  qlo = qlo < 0 ? 0 : (qlo > L - 1 ? L - 1 : qlo);
  int qhi = qlo + 1;
  qhi = qhi > L - 1 ? L - 1 : qhi;

  const float pa = warp_pos((float)qlo, sf, nlf);
  const float pb = warp_pos((float)qhi, sf, nlf);
  const float pafl = floorf(pa);
  const float pbfl = floorf(pb);
  const float fa = pa - pafl;
  const float fb = pb - pbfl;
  int la = (int)pafl, lb = (int)pbfl;
  int ha = la + 1 > kT - 1 ? kT - 1 : la + 1;
  int hb = lb + 1 > kT - 1 ? kT - 1 : lb + 1;

  auto rel = [&](int v) {
    int r = v - lo_min;
    return r < 0 ? 0 : (r > count - 1 ? count - 1 : r);
  };
  const int ra = rel(la), rha = rel(ha), rb2 = rel(lb), rhb = rel(hb);

  const float ca = 1.0f - fa, cb = 1.0f - fb, cq = 1.0f - qf;
  float* orow = out + (size_t)b * kC * kT + t;
#pragma unroll
  for (int c = 0; c < kC; ++c) {
    const float* r = &sbuf[c * kSrcStride];
    const float va = r[ra] * ca + r[rha] * fa;   // warped[qlo]
    const float vb = r[rb2] * cb + r[rhb] * fb;  // warped[qhi]
    orow[(size_t)c * kT] = va * cq + vb * qf;    // stage-2 lerp
  }
}

extern "C" void kernel_launch(void* const* d_in, const int* in_sizes, int n_in,
                              void* d_out, int out_size, void* d_ws,
                              size_t ws_size, hipStream_t stream) {
  (void)in_sizes;
  (void)n_in;
  (void)out_size;
  (void)d_ws;
  (void)ws_size;
  const float* x = (const float*)d_in[0];
  const int* ws = (const int*)d_in[1];
  const int* nl = (const int*)d_in[2];
  float* out = (float*)d_out;

  dim3 grid(kB * kTilesPerB);  // 8192 blocks
  dim3 block(kTile);           // 256 threads = 8 wave32
  hipLaunchKernelGGL(window_warp_kernel, grid, block, 0, stream, x, ws, nl,
                     out);
}